// BasePBC_54451595378994
// MI455X (gfx1250) — compile-verified
//
#include <hip/hip_runtime.h>
#include <stdint.h>
#include <stddef.h>

// ---------------- problem constants (fixed by the reference) ----------------
#define NTAPS   449       // |m*n| <= 25, |m|,|n| <= 25
#define LHALF   25
#define HALO    50        // max |m+n|
#define TILE    64        // w-positions per block (divides W=16384)
#define FILLN   (TILE + 2 * HALO)   // 164 LDS rows (float2 per row per re/im)
#define THREADS 256

// ---------------- compile-time tap table (matches Python m-outer/n-inner order) ----
constexpr int tap_count() {
  int c = 0;
  for (int m = -LHALF; m <= LHALF; ++m)
    for (int n = -LHALF; n <= LHALF; ++n) {
      int p = m * n; if (p < 0) p = -p;
      if (p <= LHALF) ++c;
    }
  return c;
}
static_assert(tap_count() == NTAPS, "tap enumeration mismatch vs reference");

struct TapPack { uint32_t pk[NTAPS]; };

constexpr TapPack make_taps() {
  TapPack t{};
  int c = 0;
  for (int m = -LHALF; m <= LHALF; ++m)
    for (int n = -LHALF; n <= LHALF; ++n) {
      int p = m * n; if (p < 0) p = -p;
      if (p <= LHALF) {
        uint32_t jn  = (uint32_t)(HALO - n);        // E[w-n]      -> sBuf[w + jn]
        uint32_t jmn = (uint32_t)(HALO - (m + n));  // E[w-m-n]    -> sBuf[w + jmn]
        uint32_t jm  = (uint32_t)(HALO - m);        // E[w-m]      -> sBuf[w + jm]
        t.pk[c++] = jn | (jmn << 8) | (jm << 16);
      }
    }
  return t;
}
__constant__ TapPack TAPS = make_taps();

// ---------------- gfx1250 async global->LDS path (guarded) ----------------
#if defined(__has_builtin)
#  if __has_builtin(__builtin_amdgcn_global_load_async_to_lds_b64)
#    define USE_ASYNC_LDS 1
#  endif
#endif
#ifndef USE_ASYNC_LDS
#  define USE_ASYNC_LDS 0
#endif

typedef __attribute__((ext_vector_type(2))) float vf2;
typedef int v2i __attribute__((vector_size(8)));   // matches builtin param type

__global__ __launch_bounds__(THREADS) void BasePBC_54451595378994_kernel(
    const float* __restrict__ Ere, const float* __restrict__ Eim,
    vf2* __restrict__ out, int W, int wmask)
{
  __shared__ vf2 sRe[FILLN];   // sRe[j] = {Re E[.,0], Re E[.,1]}
  __shared__ vf2 sIm[FILLN];

  const int b  = blockIdx.y;                 // batch (grid.y = B)
  const int w0 = blockIdx.x * TILE;          // tile start (grid.x = W/TILE)

  // ---- stage E tile (+halo, wrapped mod W) into LDS ----
  const vf2* gRe = (const vf2*)Ere + (size_t)b * W;   // one float2 (2 modes) per w
  const vf2* gIm = (const vf2*)Eim + (size_t)b * W;
  for (int j = threadIdx.x; j < FILLN; j += THREADS) {
    const int wsrc = (w0 - HALO + j) & wmask;         // W power-of-two wrap
#if USE_ASYNC_LDS
    __builtin_amdgcn_global_load_async_to_lds_b64(
        (v2i*)(gRe + wsrc), (v2i*)&sRe[j], 0, 0);
    __builtin_amdgcn_global_load_async_to_lds_b64(
        (v2i*)(gIm + wsrc), (v2i*)&sIm[j], 0, 0);
#else
    sRe[j] = gRe[wsrc];
    sIm[j] = gIm[wsrc];
#endif
  }
#if USE_ASYNC_LDS
#  if __has_builtin(__builtin_amdgcn_s_wait_asynccnt)
  __builtin_amdgcn_s_wait_asynccnt(0);
#  else
  asm volatile("s_wait_asynccnt 0x0" ::: "memory");
#  endif
#endif
  __syncthreads();

  // ---- compute: lanes over taps s (coalesced stores), loop over w ----
  const size_t obase = (size_t)(b * W + w0) * (size_t)(2 * NTAPS);
  for (int s = threadIdx.x; s < NTAPS; s += THREADS) {
    const uint32_t pk = TAPS.pk[s];
    const int jn  =  pk        & 0xFF;
    const int jmn = (pk >>  8) & 0xFF;
    const int jm  = (pk >> 16) & 0xFF;
    vf2* o0 = out + obase + (size_t)s;          // (b, w0, p=0, s)
    vf2* o1 = o0 + NTAPS;                        // (b, w0, p=1, s)
#pragma unroll 8
    for (int w = 0; w < TILE; ++w) {
      const vf2 nre  = sRe[w + jn],  nim  = sIm[w + jn];   // E[w-n]
      const vf2 mnre = sRe[w + jmn], mnim = sIm[w + jmn];  // E[w-m-n]
      const vf2 mre  = sRe[w + jm],  mim  = sIm[w + jm];   // E[w-m]
      // A = sum over modes of En * conj(Emn)   (same for both output modes)
      const float Ar = nre.x * mnre.x + nim.x * mnim.x
                     + nre.y * mnre.y + nim.y * mnim.y;
      const float Ai = nim.x * mnre.x - nre.x * mnim.x
                     + nim.y * mnre.y - nre.y * mnim.y;
      // F_p = A * Em_p
      const vf2 F0 = { Ar * mre.x - Ai * mim.x, Ar * mim.x + Ai * mre.x };
      const vf2 F1 = { Ar * mre.y - Ai * mim.y, Ar * mim.y + Ai * mre.y };
      __builtin_nontemporal_store(F0, o0);      // 235 MB stream: NT hint, > L2
      __builtin_nontemporal_store(F1, o1);
      o0 += 2 * NTAPS;
      o1 += 2 * NTAPS;
    }
  }
}

extern "C" void kernel_launch(void* const* d_in, const int* in_sizes, int n_in,
                              void* d_out, int out_size, void* d_ws, size_t ws_size,
                              hipStream_t stream) {
  (void)n_in; (void)out_size; (void)d_ws; (void)ws_size;
  const float* Ere = (const float*)d_in[0];
  const float* Eim = (const float*)d_in[1];
  const int B = 2, NMODES = 2;
  const int W = in_sizes[0] / (B * NMODES);   // 16384 (power of two)
  dim3 grid(W / TILE, B), block(THREADS);
  hipLaunchKernelGGL(BasePBC_54451595378994_kernel, grid, block, 0, stream,
                     Ere, Eim, (vf2*)d_out, W, W - 1);
}